// GRU_69630009803243
// MI455X (gfx1250) — compile-verified
//
#include <hip/hip_runtime.h>
#include <math.h>

// Problem constants (match reference)
#define T_STEPS 16384
#define IN_DIM  64
#define HID     1024
#define G3      3072   // 3*HID
#define OUT_DIM 2

#define NBLK 32        // persistent recurrence blocks (co-resident)
#define NTHR 256       // 8 wave32 per block

typedef float v2f __attribute__((ext_vector_type(2)));
typedef float v8f __attribute__((ext_vector_type(8)));

// ---------------- kernel 0: per-call init (h buffers + barrier) ----------------
__global__ void GRU_init_kernel(float* __restrict__ hbuf, unsigned int* __restrict__ bar) {
  int i = threadIdx.x;
  if (i == 0) *bar = 0u;
  for (int k = i; k < 2 * HID; k += blockDim.x) hbuf[k] = 0.f;
}

// ---------------- kernel 1: gi = input @ W_ih^T + b_ih via fp32 WMMA ----------------
// One wave per 16x16 output tile; K=64 as 16 x (16x16x4) fp32 WMMAs (exact fp32).
__global__ void __launch_bounds__(256)
GRU_gi_wmma_kernel(const float* __restrict__ x,     // [T, 64]
                   const float* __restrict__ w_ih,  // [3072, 64]
                   const float* __restrict__ b_ih,  // [3072]
                   float* __restrict__ gi)          // [T, 3072]
{
  const int tid  = threadIdx.x;
  const int lane = tid & 31;
  const int wave = tid >> 5;
  const int gw   = blockIdx.x * 8 + wave;          // global wave id
  const int tileM = gw / (G3 / 16);
  const int tileN = gw % (G3 / 16);
  const int t0 = tileM * 16;
  const int n0 = tileN * 16;

  const int half = lane >> 4;                      // 0: K+{0,1}, 1: K+{2,3}
  const int l16  = lane & 15;
  const int koff = half * 2;

  const float* __restrict__ xrow = x    + (size_t)(t0 + l16) * IN_DIM + koff; // A: M=l16
  const float* __restrict__ wrow = w_ih + (size_t)(n0 + l16) * IN_DIM + koff; // B: N=l16

  v8f c = {};
#pragma unroll
  for (int k0 = 0; k0 < IN_DIM; k0 += 4) {
    v2f a, b;
    a.x = xrow[k0];     a.y = xrow[k0 + 1];
    b.x = wrow[k0];     b.y = wrow[k0 + 1];
    // D = A(16x4) * B(4x16) + C, fp32 exact
    c = __builtin_amdgcn_wmma_f32_16x16x4_f32(false, a, false, b, (short)0, c, false, false);
  }

  // C layout: VGPR r -> (M = r + half*8, N = l16)
  const float bv = b_ih[n0 + l16];
  const int mbase = t0 + half * 8;
  float* __restrict__ outp = gi + n0 + l16;
#pragma unroll
  for (int r = 0; r < 8; ++r)
    outp[(size_t)(mbase + r) * G3] = c[r] + bv;
}

// ---------------- kernel 2: persistent sequential GRU scan ----------------
// 32 blocks x 8 waves; each wave owns 4 hidden indices j (computes rows j, H+j, 2H+j).
// h staged in LDS each step; grid-wide sync via monotonic atomic barrier.
__global__ void __launch_bounds__(NTHR)
GRU_scan_kernel(const float* __restrict__ w_hh,  // [3072, 1024]
                const float* __restrict__ b_hh,  // [3072]
                const float* __restrict__ gi,    // [T, 3072]
                float* __restrict__ hbuf,        // [2][1024]
                unsigned int* __restrict__ bar)
{
  __shared__ float hs[HID];
  const int tid  = threadIdx.x;
  const int lane = tid & 31;
  const int wave = tid >> 5;
  const int jbase = (blockIdx.x * 8 + wave) * 4;   // 4 j per wave (32*8*4 = 1024)

  int cur = 0;
  for (int t = 0; t < T_STEPS; ++t) {
    const float* __restrict__ hc = hbuf + cur * HID;
    float*       __restrict__ hn = hbuf + (cur ^ 1) * HID;

    // stage h into LDS
    for (int k = tid; k < HID; k += NTHR) hs[k] = hc[k];
    __syncthreads();

    const float* __restrict__ git = gi + (size_t)t * G3;
    // software prefetch of next step's gi row
    if (t + 1 < T_STEPS && wave == 0)
      __builtin_prefetch((const void*)(git + G3 + lane * (G3 / 32)), 0, 1);

#pragma unroll
    for (int jj = 0; jj < 4; ++jj) {
      const int j = jbase + jj;
      const float* __restrict__ wr = w_hh + (size_t)j * HID;
      const float* __restrict__ wz = w_hh + (size_t)(HID + j) * HID;
      const float* __restrict__ wn = w_hh + (size_t)(2 * HID + j) * HID;

      float ar = 0.f, az = 0.f, an = 0.f;
#pragma unroll
      for (int i = 0; i < HID / 128; ++i) {       // 8 iters, float4 per lane
        const int k4 = 4 * lane + 128 * i;
        const float4 h4 = *(const float4*)&hs[k4];
        const float4 r4 = *(const float4*)&wr[k4];
        const float4 z4 = *(const float4*)&wz[k4];
        const float4 n4 = *(const float4*)&wn[k4];
        ar = fmaf(r4.x, h4.x, fmaf(r4.y, h4.y, fmaf(r4.z, h4.z, fmaf(r4.w, h4.w, ar))));
        az = fmaf(z4.x, h4.x, fmaf(z4.y, h4.y, fmaf(z4.z, h4.z, fmaf(z4.w, h4.w, az))));
        an = fmaf(n4.x, h4.x, fmaf(n4.y, h4.y, fmaf(n4.z, h4.z, fmaf(n4.w, h4.w, an))));
      }
#pragma unroll
      for (int off = 16; off; off >>= 1) {        // wave32 xor-reduce
        ar += __shfl_xor(ar, off, 32);
        az += __shfl_xor(az, off, 32);
        an += __shfl_xor(an, off, 32);
      }
      if (lane == 0) {
        const float gr = ar + b_hh[j];
        const float gz = az + b_hh[HID + j];
        const float gn = an + b_hh[2 * HID + j];
        const float r  = 1.f / (1.f + expf(-(git[j] + gr)));
        const float z  = 1.f / (1.f + expf(-(git[HID + j] + gz)));
        const float n  = tanhf(git[2 * HID + j] + r * gn);
        hn[j] = (1.f - z) * n + z * hs[j];
      }
    }
    __syncthreads();                               // all hs reads / hn stores done

    if (tid == 0) {                                // grid barrier (monotonic counter)
      __threadfence();
      __hip_atomic_fetch_add(bar, 1u, __ATOMIC_RELEASE, __HIP_MEMORY_SCOPE_AGENT);
      const unsigned int target = (unsigned int)(t + 1) * NBLK;
      while (__hip_atomic_load(bar, __ATOMIC_ACQUIRE, __HIP_MEMORY_SCOPE_AGENT) < target)
        __builtin_amdgcn_s_sleep(1);
      __threadfence();
    }
    __syncthreads();
    cur ^= 1;
  }
}

// ---------------- kernel 3: logits + log_softmax ----------------
__global__ void __launch_bounds__(64)
GRU_head_kernel(const float* __restrict__ hT,    // [1024]
                const float* __restrict__ fc_w,  // [2, 1024]
                const float* __restrict__ fc_b,  // [2]
                float* __restrict__ out)         // [2]
{
  __shared__ float logits[OUT_DIM];
  const int lane = threadIdx.x & 31;
  const int wave = threadIdx.x >> 5;             // wave = output index
  const float* __restrict__ w = fc_w + wave * HID;
  float acc = 0.f;
#pragma unroll
  for (int i = 0; i < HID / 32; ++i) {
    const int k = lane + 32 * i;
    acc = fmaf(w[k], hT[k], acc);
  }
#pragma unroll
  for (int off = 16; off; off >>= 1) acc += __shfl_xor(acc, off, 32);
  if (lane == 0) logits[wave] = acc + fc_b[wave];
  __syncthreads();
  if (threadIdx.x == 0) {
    const float l0 = logits[0], l1 = logits[1];
    const float m  = fmaxf(l0, l1);
    const float lse = m + logf(expf(l0 - m) + expf(l1 - m));
    out[0] = l0 - lse;
    out[1] = l1 - lse;
  }
}

extern "C" void kernel_launch(void* const* d_in, const int* in_sizes, int n_in,
                              void* d_out, int out_size, void* d_ws, size_t ws_size,
                              hipStream_t stream) {
  const float* x    = (const float*)d_in[0];  // [T,1,64]
  const float* w_ih = (const float*)d_in[1];  // [3072,64]
  const float* w_hh = (const float*)d_in[2];  // [3072,1024]
  const float* b_ih = (const float*)d_in[3];  // [3072]
  const float* b_hh = (const float*)d_in[4];  // [3072]
  const float* fc_w = (const float*)d_in[5];  // [2,1024]
  const float* fc_b = (const float*)d_in[6];  // [2]
  float* out = (float*)d_out;

  // workspace layout: [barrier (16 floats pad)] [h double buffer 2*1024] [gi T*3072]
  float* ws = (float*)d_ws;
  unsigned int* bar = (unsigned int*)ws;
  float* hbuf = ws + 16;
  float* gi   = ws + 16 + 2 * HID;           // needs ~201.3 MB of d_ws

  GRU_init_kernel<<<1, 256, 0, stream>>>(hbuf, bar);

  const int waves  = (T_STEPS / 16) * (G3 / 16);   // 196608 tiles
  GRU_gi_wmma_kernel<<<waves / 8, 256, 0, stream>>>(x, w_ih, b_ih, gi);

  GRU_scan_kernel<<<NBLK, NTHR, 0, stream>>>(w_hh, b_hh, gi, hbuf, bar);

  // final h lives in buffer (T & 1) — T even -> buffer 0
  GRU_head_kernel<<<1, 64, 0, stream>>>(hbuf + (T_STEPS & 1) * HID, fc_w, fc_b, out);
}